// HiearchSoft_48619029790893
// MI455X (gfx1250) — compile-verified
//
#include <hip/hip_runtime.h>
#include <hip/hip_bf16.h>
#include <math.h>

// CDNA5 (gfx1250) WMMA types
typedef float v2f __attribute__((ext_vector_type(2)));
typedef float v8f __attribute__((ext_vector_type(8)));

#define D_DIM 128            // embedding dim (K)
#define LDS_STRIDE 132       // 128 + 4 floats pad -> conflict-free ds_load_b64 fragments
#define ROWS_PER_WAVE 16
#define WAVES_PER_BLOCK 8
#define ROWS_PER_BLOCK (ROWS_PER_WAVE * WAVES_PER_BLOCK)

__device__ __forceinline__ float log_sigmoid_f(float t) {
    // log(sigmoid(t)) = min(t,0) - log(1 + exp(-|t|)), arg of log in (1,2]
    return fminf(t, 0.0f) - __logf(1.0f + __expf(-fabsf(t)));
}

__global__ __launch_bounds__(256)
void hs_wmma_kernel(const float* __restrict__ xw,
                    const float* __restrict__ W,
                    const float* __restrict__ h_code,
                    const int*   __restrict__ h_path,
                    float* __restrict__ out,
                    int B, int L) {
    __shared__ float ldsW[32 * LDS_STRIDE];   // gathered+padded Wp: 32 x 128 (rows >= L zeroed)

    const int tid  = threadIdx.x;
    const int lane = tid & 31;                // wave32
    const int wave = tid >> 5;

    // ---- Stage the L gathered W rows into LDS (pad to 32 rows with zeros) ----
    for (int idx = tid; idx < 32 * D_DIM; idx += 256) {
        int n = idx >> 7;               // row (path node)
        int k = idx & (D_DIM - 1);      // column
        float v = 0.0f;
        if (n < L) v = W[(size_t)h_path[n] * D_DIM + k];
        ldsW[n * LDS_STRIDE + k] = v;
    }
    __syncthreads();

    const int rowBase = blockIdx.x * ROWS_PER_BLOCK + wave * ROWS_PER_WAVE;
    const int mlo   = lane & 15;        // M (for A) / N (for B) within half-wave
    const int khalf = (lane >> 4) * 2;  // upper half-wave holds K+2,K+3 of each K=4 step

    int arow = rowBase + mlo;
    if (arow >= B) arow = B - 1;        // clamp loads; stores predicated below
    const float* aPtr  = xw + (size_t)arow * D_DIM + khalf;
    const float* b0Ptr = &ldsW[ mlo       * LDS_STRIDE + khalf];  // columns n = 0..15
    const float* b1Ptr = &ldsW[(mlo + 16) * LDS_STRIDE + khalf];  // columns n = 16..31

    v8f c0 = {};   // z tile, N = 0..15
    v8f c1 = {};   // z tile, N = 16..31 (padded)

    #pragma unroll
    for (int k = 0; k < D_DIM; k += 4) {
        v2f a  = *(const v2f*)(aPtr  + k);   // A fragment: 16x4 fp32
        v2f b0 = *(const v2f*)(b0Ptr + k);   // B fragment: 4x16 fp32 (LDS)
        v2f b1 = *(const v2f*)(b1Ptr + k);
        c0 = __builtin_amdgcn_wmma_f32_16x16x4_f32(false, a, false, b0,
                                                   (short)0, c0, false, false);
        c1 = __builtin_amdgcn_wmma_f32_16x16x4_f32(false, a, false, b1,
                                                   (short)0, c1, false, false);
    }

    // ---- epilogue: p = log_sigmoid((2*code-1) * z), masked sum over valid columns ----
    const bool  val0 = (mlo       < L);
    const bool  val1 = ((16 + mlo) < L);
    const float s0 = val0 ? (2.0f * h_code[mlo]      - 1.0f) : 0.0f;
    const float s1 = val1 ? (2.0f * h_code[16 + mlo] - 1.0f) : 0.0f;

    #pragma unroll
    for (int j = 0; j < 8; ++j) {
        // VGPR j holds row M=j (lanes 0-15) / M=8+j (lanes 16-31), column N = mlo
        float p = 0.0f;
        if (val0) p += log_sigmoid_f(s0 * c0[j]);
        if (val1) p += log_sigmoid_f(s1 * c1[j]);
        // reduce the 16 columns spread across each 16-lane half
        p += __shfl_xor(p, 1, 32);
        p += __shfl_xor(p, 2, 32);
        p += __shfl_xor(p, 4, 32);
        p += __shfl_xor(p, 8, 32);
        const int row = rowBase + ((lane >> 4) << 3) + j;  // lane0 -> j, lane16 -> 8+j
        if (mlo == 0 && row < B) out[row] = p;
    }
}

extern "C" void kernel_launch(void* const* d_in, const int* in_sizes, int n_in,
                              void* d_out, int out_size, void* d_ws, size_t ws_size,
                              hipStream_t stream) {
    const float* xw     = (const float*)d_in[0];   // (B, 128) fp32
    const float* W      = (const float*)d_in[1];   // (N, 128) fp32
    const float* h_code = (const float*)d_in[2];   // (L,)     fp32
    const int*   h_path = (const int*)  d_in[3];   // (L,)     int32
    float* out = (float*)d_out;                    // (B, 1)   fp32

    const int B = in_sizes[0] / D_DIM;
    const int L = in_sizes[2];                     // 20; must be <= 32

    const int grid = (B + ROWS_PER_BLOCK - 1) / ROWS_PER_BLOCK;
    hs_wmma_kernel<<<grid, 256, 0, stream>>>(xw, W, h_code, h_path, out, B, L);
}